// DataNet_39049842655641
// MI455X (gfx1250) — compile-verified
//
#include <hip/hip_runtime.h>
#include <hip/hip_bf16.h>

typedef __attribute__((ext_vector_type(2))) float v2f;
typedef __attribute__((ext_vector_type(8))) float v8f;

#define PI2F 6.28318530717958647692f

// ---------------------------------------------------------------------------
// Async global -> LDS staging (CDNA5 Tensor/Async data path, ASYNCcnt-tracked).
// Inline asm keeps this portable across ROCm 7.2 / amdgpu-toolchain builtins.
// LDS destination offset = low 32 bits of the flat pointer (ISA 10.2);
// global source is a 64-bit VGPR pair, GV addressing mode.
// ---------------------------------------------------------------------------
__device__ __forceinline__ void async_ld_b64(void* lds_dst, const void* gsrc) {
  unsigned ldsoff = (unsigned)(size_t)lds_dst;
  asm volatile("global_load_async_to_lds_b64 %0, %1, off"
               :: "v"(ldsoff), "v"(gsrc) : "memory");
}
__device__ __forceinline__ void async_ld_b128(void* lds_dst, const void* gsrc) {
  unsigned ldsoff = (unsigned)(size_t)lds_dst;
  asm volatile("global_load_async_to_lds_b128 %0, %1, off"
               :: "v"(ldsoff), "v"(gsrc) : "memory");
}
__device__ __forceinline__ void wait_async0() {
  asm volatile("s_wait_asynccnt 0x0" ::: "memory");
}

// ---------------------------------------------------------------------------
// Complex 16x16x16 GEMM tile on the fp32 WMMA path:
//   D = F * B  with F = Ar + i*Ai (DFT-16 matrix, held in A-operand regs),
//   B = Br + i*Bi.  F32 WMMA has no A/B negate (ISA: NEG = CNeg only), so we
//   carry mai = -Ai.  16 x v_wmma_f32_16x16x4_f32 per complex tile.
// ---------------------------------------------------------------------------
__device__ __forceinline__ void cwmma16(const v2f ar[4], const v2f ai[4],
                                        const v2f mai[4],
                                        const v2f br[4], const v2f bi[4],
                                        v8f& dr, v8f& di) {
#pragma unroll
  for (int c = 0; c < 4; ++c) {
    dr = __builtin_amdgcn_wmma_f32_16x16x4_f32(false, ar[c],  false, br[c], (short)0, dr, false, false);
    dr = __builtin_amdgcn_wmma_f32_16x16x4_f32(false, mai[c], false, bi[c], (short)0, dr, false, false);
    di = __builtin_amdgcn_wmma_f32_16x16x4_f32(false, ar[c],  false, bi[c], (short)0, di, false, false);
    di = __builtin_amdgcn_wmma_f32_16x16x4_f32(false, ai[c],  false, br[c], (short)0, di, false, false);
  }
}

// DFT-16 matrix F[m][k] = exp(sgn*2*pi*i*m*k/16) in WMMA A-operand layout:
// chunk c (K = 4c..4c+3): lane holds M = lane&15, K = 4c + 2*(lane>=16) + j.
__device__ __forceinline__ void dft16A(float sgn, int lane,
                                       v2f ar[4], v2f ai[4], v2f mai[4]) {
  int m  = lane & 15;
  int h2 = (lane >> 4) << 1;
#pragma unroll
  for (int c = 0; c < 4; ++c)
#pragma unroll
    for (int j = 0; j < 2; ++j) {
      int k = 4 * c + h2 + j;
      float ang = sgn * PI2F * (float)((m * k) & 15) / 16.0f;
      float s, co;
      __sincosf(ang, &s, &co);
      ar[c][j] = co; ai[c][j] = s; mai[c][j] = -s;
    }
}

// ---------------------------------------------------------------------------
// 256-point 1D FFT, one wave per transform.  N = 16*16 four-step:
//   A  = F16 * X        (X[n1][n2] = x[16 n1 + n2])
//   A' = A .* W256^{k1 n2}
//   out[k1 + 16 k2] = (F16 * A'^T)[k2][k1]
// fftshift/ifftshift folded as index ^ xin/xout on the transform axis.
// Input row staged via async global->LDS, overlapped with DFT-matrix setup.
// ---------------------------------------------------------------------------
__global__ __launch_bounds__(32) void fft256_kernel(
    const float2* __restrict__ in, float2* __restrict__ out,
    int rows_per_img, long long img_stride, long long row_stride,
    long long elem_stride, float sgn, float scale, int xin, int xout) {
  __shared__ float2 sx[256];
  __shared__ float2 sy[256];
  int lane = threadIdx.x;
  int r = blockIdx.x;
  long long base = (long long)(r / rows_per_img) * img_stride +
                   (long long)(r % rows_per_img) * row_stride;
  const float2* src = in + base;
  float2* dst = out + base;

  if (elem_stride == 1) {
    for (int p = lane; p < 128; p += 32)            // 16B/lane async copies
      async_ld_b128(&sx[2 * p], src + ((2 * p) ^ xin));
  } else {
    for (int idx = lane; idx < 256; idx += 32)      // strided column gather
      async_ld_b64(&sx[idx], src + (long long)(idx ^ xin) * elem_stride);
  }

  v2f ar[4], ai[4], mai[4];
  dft16A(sgn, lane, ar, ai, mai);                   // overlaps async copy
  int col = lane & 15;
  int h2  = (lane >> 4) << 1;

  wait_async0();
  __syncthreads();

  // stage 1: B(n1, col) = X[16 n1 + col]
  v2f br[4], bi[4];
#pragma unroll
  for (int c = 0; c < 4; ++c)
#pragma unroll
    for (int j = 0; j < 2; ++j) {
      int n = 4 * c + h2 + j;
      float2 t = sx[n * 16 + col];
      br[c][j] = t.x; bi[c][j] = t.y;
    }
  v8f dr = {}, di = {};
  cwmma16(ar, ai, mai, br, bi, dr, di);

  // twiddle W256^{m*col}, store transposed (A'^T) for stage-2 B operand
#pragma unroll
  for (int v = 0; v < 8; ++v) {
    int m = v + ((lane >> 4) << 3);
    float ang = sgn * PI2F * (float)((m * col) & 255) / 256.0f;
    float s, co; __sincosf(ang, &s, &co);
    float xr = dr[v], xi = di[v];
    sy[col * 16 + m] = make_float2(xr * co - xi * s, xr * s + xi * co);
  }
  __syncthreads();

  // stage 2: B2(n2, k1) = A'[k1][n2]
#pragma unroll
  for (int c = 0; c < 4; ++c)
#pragma unroll
    for (int j = 0; j < 2; ++j) {
      int n = 4 * c + h2 + j;
      float2 t = sy[n * 16 + col];
      br[c][j] = t.x; bi[c][j] = t.y;
    }
  v8f er = {}, ei = {};
  cwmma16(ar, ai, mai, br, bi, er, ei);

  // D element (m=k2, col=k1) -> out index k1 + 16*k2
#pragma unroll
  for (int v = 0; v < 8; ++v) {
    int m = v + ((lane >> 4) << 3);
    int g = col + 16 * m;
    float2 o; o.x = er[v] * scale; o.y = ei[v] * scale;
    dst[(long long)(g ^ xout) * elem_stride] = o;
  }
}

// ---------------------------------------------------------------------------
// 1024-point 1D FFT, one wave per transform.  N = 16*64, 64 = 16*4:
//   stage1: A = F16 * X (16x64), twiddle W1024^{k1 n2}
//   stage2: per (k1,q): DFT16 over p of A'[k1][4p+q], twiddle W64^{q j1}
//   stage3: radix-4 over q (VALU), k = k1 + 16 j1 + 256 j2
// ---------------------------------------------------------------------------
__global__ __launch_bounds__(32) void fft1024_kernel(
    const float2* __restrict__ in, float2* __restrict__ out,
    int rows_per_img, long long img_stride, long long row_stride,
    long long elem_stride, float sgn, float scale, int xin, int xout) {
  __shared__ float2 sx[1024];
  __shared__ float2 sy[1024];
  int lane = threadIdx.x;
  int r = blockIdx.x;
  long long base = (long long)(r / rows_per_img) * img_stride +
                   (long long)(r % rows_per_img) * row_stride;
  const float2* src = in + base;
  float2* dst = out + base;

  if (elem_stride == 1) {
    for (int p = lane; p < 512; p += 32)
      async_ld_b128(&sx[2 * p], src + ((2 * p) ^ xin));
  } else {
    for (int idx = lane; idx < 1024; idx += 32)
      async_ld_b64(&sx[idx], src + (long long)(idx ^ xin) * elem_stride);
  }

  v2f ar[4], ai[4], mai[4];
  dft16A(sgn, lane, ar, ai, mai);                   // overlaps async copy
  int col = lane & 15;
  int h2  = (lane >> 4) << 1;

  wait_async0();
  __syncthreads();

  // stage 1: 4 column tiles over n2
  for (int t = 0; t < 4; ++t) {
    int n2 = 16 * t + col;
    v2f br[4], bi[4];
#pragma unroll
    for (int c = 0; c < 4; ++c)
#pragma unroll
      for (int j = 0; j < 2; ++j) {
        int n1 = 4 * c + h2 + j;
        float2 tv = sx[n1 * 64 + n2];
        br[c][j] = tv.x; bi[c][j] = tv.y;
      }
    v8f dr = {}, di = {};
    cwmma16(ar, ai, mai, br, bi, dr, di);
#pragma unroll
    for (int v = 0; v < 8; ++v) {
      int m = v + ((lane >> 4) << 3);   // k1
      float ang = sgn * PI2F * (float)((m * n2) & 1023) / 1024.0f;
      float s, co; __sincosf(ang, &s, &co);
      float xr = dr[v], xi = di[v];
      sy[m * 64 + n2] = make_float2(xr * co - xi * s, xr * s + xi * co);
    }
  }
  __syncthreads();

  // stage 2: 4 tiles over u = (k1,q) pairs; B(p,u) = A'[k1][4p+q]
  for (int t = 0; t < 4; ++t) {
    int u  = 16 * t + col;
    int k1 = u >> 2, q = u & 3;
    v2f br[4], bi[4];
#pragma unroll
    for (int c = 0; c < 4; ++c)
#pragma unroll
      for (int j = 0; j < 2; ++j) {
        int p = 4 * c + h2 + j;
        float2 tv = sy[k1 * 64 + 4 * p + q];
        br[c][j] = tv.x; bi[c][j] = tv.y;
      }
    v8f dr = {}, di = {};
    cwmma16(ar, ai, mai, br, bi, dr, di);
#pragma unroll
    for (int v = 0; v < 8; ++v) {
      int m = v + ((lane >> 4) << 3);   // j1
      float ang = sgn * PI2F * (float)((q * m) & 63) / 64.0f;
      float s, co; __sincosf(ang, &s, &co);
      float xr = dr[v], xi = di[v];
      sx[k1 * 64 + q * 16 + m] = make_float2(xr * co - xi * s, xr * s + xi * co);
    }
  }
  __syncthreads();

  // stage 3: radix-4 butterflies over q, write out
  for (int v = lane; v < 256; v += 32) {
    int k1 = v >> 4, j1 = v & 15;
    int b0 = k1 * 64 + j1;
    float2 g0 = sx[b0], g1 = sx[b0 + 16], g2 = sx[b0 + 32], g3 = sx[b0 + 48];
    float Axr = g0.x + g2.x, Axi = g0.y + g2.y;
    float Bxr = g0.x - g2.x, Bxi = g0.y - g2.y;
    float Cxr = g1.x + g3.x, Cxi = g1.y + g3.y;
    float Dxr = g1.x - g3.x, Dxi = g1.y - g3.y;
    float Exr = -sgn * Dxi, Exi = sgn * Dxr;  // i*sgn*D
    int kb = k1 + 16 * j1;
    float2 o;
    o.x = (Axr + Cxr) * scale; o.y = (Axi + Cxi) * scale;
    dst[(long long)(kb         ^ xout) * elem_stride] = o;
    o.x = (Bxr + Exr) * scale; o.y = (Bxi + Exi) * scale;
    dst[(long long)((kb + 256) ^ xout) * elem_stride] = o;
    o.x = (Axr - Cxr) * scale; o.y = (Axi - Cxi) * scale;
    dst[(long long)((kb + 512) ^ xout) * elem_stride] = o;
    o.x = (Bxr - Exr) * scale; o.y = (Bxi - Exi) * scale;
    dst[(long long)((kb + 768) ^ xout) * elem_stride] = o;
  }
}

// ---------------------------------------------------------------------------
// Elementwise / data-movement kernels
// ---------------------------------------------------------------------------
__global__ void pack_kernel(const float* __restrict__ re,
                            const float* __restrict__ im,
                            float2* __restrict__ out) {
  long long g = (long long)blockIdx.x * blockDim.x + threadIdx.x;
  out[g] = make_float2(re[g], im[g]);
}

__global__ void zero_kernel(float* __restrict__ p, long long n) {
  long long g = (long long)blockIdx.x * blockDim.x + threadIdx.x;
  if (g < n) p[g] = 0.0f;
}

// gather z_f patches * CTF, with the patch ifftshift folded into the store
__global__ void gather_kernel(const float2* __restrict__ zf,
                              const int* __restrict__ masks,
                              const float* __restrict__ ctf,
                              float2* __restrict__ xx) {
  long long gid = (long long)blockIdx.x * blockDim.x + threadIdx.x;
  int j = (int)(gid & 255), i = (int)((gid >> 8) & 255);
  int l = (int)((gid >> 16) & 63), b = (int)(gid >> 22);
  int mr = masks[2 * l] - 1, mc = masks[2 * l + 1] - 1;
  float2 v = zf[(long long)b * 1048576 + (long long)(mr + i) * 1024 + (mc + j)];
  float c = ctf[i * 256 + j];
  xx[((long long)(b * 64 + l) << 16) + ((i ^ 128) << 8) + (j ^ 128)] =
      make_float2(v.x * c, v.y * c);
}

// Cz = Bz - sqrt(Y) * Bz/|Bz|   (angle(0)=0 -> phase (1,0))
__global__ void phase_kernel(const float2* __restrict__ bz,
                             const float* __restrict__ Y,
                             float2* __restrict__ cz) {
  long long g = (long long)blockIdx.x * blockDim.x + threadIdx.x;
  float2 z = bz[g];
  float sy = sqrtf(Y[g]);
  float mag = sqrtf(z.x * z.x + z.y * z.y);
  float pr = 1.0f, pi = 0.0f;
  if (mag > 0.0f) { pr = z.x / mag; pi = z.y / mag; }
  cz[g] = make_float2(z.x - sy * pr, z.y - sy * pi);
}

// S[b, mr+i, mc+j] += fftshift(FZ)[i][j] * ctf / L   (masks overlap -> atomics)
__global__ void scatter_kernel(const float2* __restrict__ fz,
                               const int* __restrict__ masks,
                               const float* __restrict__ ctf,
                               float* __restrict__ S) {
  long long gid = (long long)blockIdx.x * blockDim.x + threadIdx.x;
  int j = (int)(gid & 255), i = (int)((gid >> 8) & 255);
  int l = (int)((gid >> 16) & 63), b = (int)(gid >> 22);
  int mr = masks[2 * l] - 1, mc = masks[2 * l + 1] - 1;
  float2 v = fz[((long long)(b * 64 + l) << 16) + ((i ^ 128) << 8) + (j ^ 128)];
  float c = ctf[i * 256 + j] * (1.0f / 64.0f);
  long long d = ((long long)b << 20) + (long long)(mr + i) * 1024 + (mc + j);
  atomicAdd(&S[2 * d],     v.x * c);
  atomicAdd(&S[2 * d + 1], v.y * c);
}

// im_rc = (1-100*eta*lamb)*Img_rnew + 100*eta*lamb*temp_x - 10*eta*W
__global__ void combine_kernel(const float* __restrict__ Ia,
                               const float* __restrict__ Cr,
                               const float* __restrict__ Ci,
                               const float2* __restrict__ W,
                               const float* __restrict__ lamb,
                               const float* __restrict__ eta,
                               float* __restrict__ out) {
  long long g = (long long)blockIdx.x * blockDim.x + threadIdx.x;
  float e = eta[0], la = lamb[0];
  float k2 = 100.0f * e * la;
  float k1 = 1.0f - k2;
  float k3 = 10.0f * e;
  float cr = Cr[g], ci = Ci[g], a = Ia[g];
  float mag = sqrtf(cr * cr + ci * ci);
  float s = a / (mag + 1e-6f);
  float2 w = W[g];
  float rr = k1 * cr + k2 * cr * s - k3 * w.x;
  float ri = k1 * ci + k2 * ci * s - k3 * w.y;
  out[g] = sqrtf(rr * rr + ri * ri);           // im_ra
  out[4194304 + 2 * g]     = rr;               // im_rc interleaved
  out[4194304 + 2 * g + 1] = ri;
}

// ---------------------------------------------------------------------------
extern "C" void kernel_launch(void* const* d_in, const int* in_sizes, int n_in,
                              void* d_out, int out_size, void* d_ws, size_t ws_size,
                              hipStream_t stream) {
  const float* Img_a = (const float*)d_in[0];
  const float* Cr    = (const float*)d_in[1];
  const float* Ci    = (const float*)d_in[2];
  const float* Y     = (const float*)d_in[3];
  const int*   Masks = (const int*)d_in[4];
  const float* CTF   = (const float*)d_in[5];
  const float* lamb  = (const float*)d_in[6];
  const float* eta1  = (const float*)d_in[7];
  float* out = (float*)d_out;
  (void)in_sizes; (void)n_in; (void)out_size; (void)ws_size;

  float2* big0 = (float2*)d_ws;          // 4*1024*1024
  float2* big1 = big0 + 4194304;
  float2* big2 = big1 + 4194304;
  float2* pat0 = big2 + 4194304;         // 4*64*256*256
  float2* pat1 = pat0 + 16777216;

  // Img_rnew -> big0
  pack_kernel<<<16384, 256, 0, stream>>>(Cr, Ci, big0);
  // z_f = fftshift(fft2(Img_rnew)): shift = k^512 on each axis's output
  fft1024_kernel<<<4096, 32, 0, stream>>>(big0, big1, 1024, 1048576LL, 1024LL, 1LL,
                                          -1.0f, 1.0f, 0, 512);
  fft1024_kernel<<<4096, 32, 0, stream>>>(big1, big0, 1024, 1048576LL, 1LL, 1024LL,
                                          -1.0f, 1.0f, 0, 512);
  // patches: gather * CTF with ifftshift folded in
  gather_kernel<<<65536, 256, 0, stream>>>(big0, Masks, CTF, pat0);
  // Bz = ifft2(patches)
  fft256_kernel<<<65536, 32, 0, stream>>>(pat0, pat1, 256, 65536LL, 256LL, 1LL,
                                          1.0f, 1.0f / 256.0f, 0, 0);
  fft256_kernel<<<65536, 32, 0, stream>>>(pat1, pat0, 256, 65536LL, 1LL, 256LL,
                                          1.0f, 1.0f / 256.0f, 0, 0);
  // Cz = Bz - sqrt(Y)*exp(i*angle(Bz))
  phase_kernel<<<65536, 256, 0, stream>>>(pat0, Y, pat1);
  // FZ = fft2(Cz)
  fft256_kernel<<<65536, 32, 0, stream>>>(pat1, pat0, 256, 65536LL, 256LL, 1LL,
                                          -1.0f, 1.0f, 0, 0);
  fft256_kernel<<<65536, 32, 0, stream>>>(pat0, pat1, 256, 65536LL, 1LL, 256LL,
                                          -1.0f, 1.0f, 0, 0);
  // S = scatter(fftshift(FZ)*conj(CTF))/L  (CTF real)
  zero_kernel<<<32768, 256, 0, stream>>>((float*)big2, 8388608LL);
  scatter_kernel<<<65536, 256, 0, stream>>>(pat1, Masks, CTF, (float*)big2);
  // W = ifft2(ifftshift(S)): shift = idx^512 on each axis's input
  fft1024_kernel<<<4096, 32, 0, stream>>>(big2, big1, 1024, 1048576LL, 1024LL, 1LL,
                                          1.0f, 1.0f / 1024.0f, 512, 0);
  fft1024_kernel<<<4096, 32, 0, stream>>>(big1, big2, 1024, 1048576LL, 1LL, 1024LL,
                                          1.0f, 1.0f / 1024.0f, 512, 0);
  // final combine (uses inv == Img_rnew identity)
  combine_kernel<<<16384, 256, 0, stream>>>(Img_a, Cr, Ci, big2, lamb, eta1, out);
}